// PointNetSAModule_1717986918816
// MI455X (gfx1250) — compile-verified
//
#include <hip/hip_runtime.h>
#include <math.h>

// Problem constants (from reference)
#define Bn    16
#define Nn    8192
#define CINn  64
#define Sn    1024
#define Kn    32
#define TOTn  (Bn * Sn * Kn)        // 524288 columns
#define R2c   0.04f                 // RADIUS^2
#define EPSc  1e-5f
#define NXYZ  (Bn * 3 * Sn)         // 49152 floats: new_xyz portion of d_out

typedef __attribute__((ext_vector_type(2))) float v2f;
typedef __attribute__((ext_vector_type(8))) float v8f;

// ---------------------------------------------------------------------------
// Kernel 1: farthest point sampling, one block (1024 thr = 32 waves) per batch.
// Writes new_xyz (B,3,S) directly into d_out. Uses non-fused mul/add to match
// the reference's elementwise fp32 semantics, argmax tie-break = lowest index.
// ---------------------------------------------------------------------------
__global__ __launch_bounds__(1024) void fps_kernel(const float* __restrict__ xyz,
                                                   float* __restrict__ out_xyz) {
  const int b = blockIdx.x;
  const int t = threadIdx.x;
  const float* px = xyz + (size_t)b * 3 * Nn;
  const float* py = px + Nn;
  const float* pz = px + 2 * Nn;

  float lx[8], ly[8], lz[8], ld[8];
  for (int i = 0; i < 8; ++i) {
    int n = t + i * 1024;
    lx[i] = px[n]; ly[i] = py[n]; lz[i] = pz[n];
    ld[i] = 3.402823466e+38f;   // inf surrogate
  }

  __shared__ float s_v[32];
  __shared__ int   s_i[32];
  __shared__ int   s_last;

  int last = 0;
  for (int step = 0; step < Sn; ++step) {
    float cx = px[last], cy = py[last], cz = pz[last];
    if (t == 0) {
      float* o = out_xyz + (size_t)b * 3 * Sn + step;
      o[0] = cx; o[Sn] = cy; o[2 * Sn] = cz;
    }
    float bv = -1.0f; int bi = 0;
    for (int i = 0; i < 8; ++i) {
      float dx = __fsub_rn(lx[i], cx);
      float dy = __fsub_rn(ly[i], cy);
      float dz = __fsub_rn(lz[i], cz);
      float d  = __fadd_rn(__fadd_rn(__fmul_rn(dx, dx), __fmul_rn(dy, dy)),
                           __fmul_rn(dz, dz));
      float nd = fminf(ld[i], d);
      ld[i] = nd;
      int n = t + i * 1024;
      if (nd > bv || (nd == bv && n < bi)) { bv = nd; bi = n; }
    }
    // wave32 reduce (prefer larger value; tie -> smaller index)
    for (int off = 16; off > 0; off >>= 1) {
      float ov = __shfl_xor(bv, off, 32);
      int   oi = __shfl_xor(bi, off, 32);
      if (ov > bv || (ov == bv && oi < bi)) { bv = ov; bi = oi; }
    }
    int wid = t >> 5;
    if ((t & 31) == 0) { s_v[wid] = bv; s_i[wid] = bi; }
    __syncthreads();
    if (wid == 0) {
      bv = s_v[t & 31]; bi = s_i[t & 31];
      for (int off = 16; off > 0; off >>= 1) {
        float ov = __shfl_xor(bv, off, 32);
        int   oi = __shfl_xor(bi, off, 32);
        if (ov > bv || (ov == bv && oi < bi)) { bv = ov; bi = oi; }
      }
      if (t == 0) s_last = bi;
    }
    __syncthreads();
    last = s_last;
  }
}

// ---------------------------------------------------------------------------
// Kernel 2: ball query. Thread per center; first K points (index order) within
// radius; pad with first hit (center itself is always a hit, d2==0).
// xyz per batch (96KB) lives in the 192MB L2, so global reads are cheap.
// ---------------------------------------------------------------------------
__global__ __launch_bounds__(256) void ballquery_kernel(const float* __restrict__ xyz,
                                                        const float* __restrict__ newxyz,
                                                        int* __restrict__ gidx) {
  int g = blockIdx.x * 256 + threadIdx.x;   // b*S + s
  int b = g >> 10;
  int s = g & 1023;
  const float* px = xyz + (size_t)b * 3 * Nn;
  const float* py = px + Nn;
  const float* pz = px + 2 * Nn;
  const float* c  = newxyz + (size_t)b * 3 * Sn + s;
  float cx = c[0], cy = c[Sn], cz = c[2 * Sn];
  int* out = gidx + (size_t)g * Kn;
  int cnt = 0, first = 0;
  for (int n = 0; n < Nn; ++n) {
    float dx = __fsub_rn(cx, px[n]);
    float dy = __fsub_rn(cy, py[n]);
    float dz = __fsub_rn(cz, pz[n]);
    float d  = __fadd_rn(__fadd_rn(__fmul_rn(dx, dx), __fmul_rn(dy, dy)),
                         __fmul_rn(dz, dz));
    if (d < R2c) {
      if (cnt == 0) first = n;
      out[cnt] = n;
      if (++cnt == Kn) break;
    }
  }
  for (; cnt < Kn; ++cnt) out[cnt] = first;
}

// ---------------------------------------------------------------------------
// Kernel 3: build X0 (68 rows x TOT), row-major: rows 0-2 = centered xyz,
// rows 3-66 = gathered features, row 67 = zero pad (K-dim padded to mult of 4).
// ---------------------------------------------------------------------------
__global__ __launch_bounds__(256) void build_x0(const float* __restrict__ xyz,
                                                const float* __restrict__ feat,
                                                const float* __restrict__ newxyz,
                                                const int* __restrict__ gidx,
                                                float* __restrict__ X) {
  int col = blockIdx.x * 256 + threadIdx.x;
  int g = col >> 5;            // /K
  int b = g >> 10;
  int s = g & 1023;
  int n = gidx[col];
  const float* px = xyz + (size_t)b * 3 * Nn;
  const float* c  = newxyz + (size_t)b * 3 * Sn + s;
  X[(size_t)0 * TOTn + col] = px[n] - c[0];
  X[(size_t)1 * TOTn + col] = px[Nn + n] - c[Sn];
  X[(size_t)2 * TOTn + col] = px[2 * Nn + n] - c[2 * Sn];
  const float* fb = feat + (size_t)b * CINn * Nn + n;
  for (int cc = 0; cc < CINn; ++cc)
    X[(size_t)(3 + cc) * TOTn + col] = fb[(size_t)cc * Nn];
  X[(size_t)67 * TOTn + col] = 0.0f;
}

__global__ void pad_w0(const float* __restrict__ W0, float* __restrict__ Wp) {
  int i = blockIdx.x * 256 + threadIdx.x;   // 64*68 = 4352
  int o = i / 68, cc = i % 68;
  Wp[i] = (cc < 67) ? W0[o * 67 + cc] : 0.0f;
}

// ---------------------------------------------------------------------------
// fp32 WMMA 16x16x4 tile: D += A(16x4) * B(4x16), looped over the K dimension.
// Layout per ISA tables: A lane L holds row L&15, VGPR v holds K = 2*(L>>4)+v;
// B mirrors A's K striping; C/D lane L vgpr v -> (m = v+8*(L>>4), n = L&15).
// ---------------------------------------------------------------------------
__device__ inline v8f wmma_tile(const float* __restrict__ W, const float* __restrict__ X,
                                int oBase, int colBase, int CIpad, int ll, int lh) {
  v8f acc = {0.f, 0.f, 0.f, 0.f, 0.f, 0.f, 0.f, 0.f};
  const float* wrow = W + (size_t)(oBase + ll) * CIpad + 2 * lh;
  const float* xcol = X + (size_t)(2 * lh) * TOTn + colBase + ll;
  for (int kk = 0; kk < CIpad; kk += 4) {
    float2 av = *(const float2*)(wrow + kk);
    v2f a; a.x = av.x; a.y = av.y;
    v2f bb;
    bb.x = xcol[(size_t)kk * TOTn];
    bb.y = xcol[(size_t)(kk + 1) * TOTn];
    acc = __builtin_amdgcn_wmma_f32_16x16x4_f32(false, a, false, bb,
                                                (short)0, acc, false, false);
  }
  return acc;
}

// Stats pass: GEMM, reduce sum / sumsq per channel straight from accumulators.
// Each wave covers 8 column tiles -> one deterministic partial per (ch, part).
__global__ __launch_bounds__(256) void gemm_stats(const float* __restrict__ W,
                                                  const float* __restrict__ X,
                                                  int CIpad,
                                                  float* __restrict__ partS,
                                                  float* __restrict__ partQ) {
  const int lane = threadIdx.x & 31, lh = lane >> 4, ll = lane & 15;
  const int wid = blockIdx.x * 8 + (threadIdx.x >> 5);
  const int nParts = TOTn / 128;                  // 4096
  const int part = wid % nParts;
  const int oBase = (wid / nParts) * 16;
  float s1[8], s2[8];
  for (int v = 0; v < 8; ++v) { s1[v] = 0.f; s2[v] = 0.f; }
  for (int t8 = 0; t8 < 8; ++t8) {
    v8f acc = wmma_tile(W, X, oBase, part * 128 + t8 * 16, CIpad, ll, lh);
    for (int v = 0; v < 8; ++v) { s1[v] += acc[v]; s2[v] += acc[v] * acc[v]; }
  }
  for (int off = 1; off < 16; off <<= 1)
    for (int v = 0; v < 8; ++v) {
      s1[v] += __shfl_xor(s1[v], off, 32);
      s2[v] += __shfl_xor(s2[v], off, 32);
    }
  if (ll == 0)
    for (int v = 0; v < 8; ++v) {
      int ch = oBase + v + 8 * lh;
      partS[(size_t)ch * nParts + part] = s1[v];
      partQ[(size_t)ch * nParts + part] = s2[v];
    }
}

// Finalize: fixed-order reduce of partials -> per-channel scale/shift.
__global__ __launch_bounds__(256) void reduce_stats(const float* __restrict__ partS,
                                                    const float* __restrict__ partQ,
                                                    const float* __restrict__ gamma,
                                                    const float* __restrict__ beta,
                                                    float* __restrict__ scale,
                                                    float* __restrict__ shift) {
  const int ch = blockIdx.x, t = threadIdx.x;
  const int nParts = TOTn / 128;
  float s1 = 0.f, s2 = 0.f;
  for (int i = t; i < nParts; i += 256) {
    s1 += partS[(size_t)ch * nParts + i];
    s2 += partQ[(size_t)ch * nParts + i];
  }
  __shared__ float a1[256], a2[256];
  a1[t] = s1; a2[t] = s2;
  __syncthreads();
  for (int off = 128; off > 0; off >>= 1) {
    if (t < off) { a1[t] += a1[t + off]; a2[t] += a2[t + off]; }
    __syncthreads();
  }
  if (t == 0) {
    const float invN = 1.0f / (float)TOTn;
    float mean = a1[0] * invN;
    float var  = a2[0] * invN - mean * mean;
    float inv  = 1.0f / sqrtf(var + EPSc);
    float sc   = gamma[ch] * inv;
    scale[ch] = sc;
    shift[ch] = beta[ch] - mean * sc;
  }
}

// Apply pass: GEMM + normalize + ReLU -> next-layer activations.
__global__ __launch_bounds__(256) void gemm_apply(const float* __restrict__ W,
                                                  const float* __restrict__ X,
                                                  int CIpad,
                                                  const float* __restrict__ scale,
                                                  const float* __restrict__ shift,
                                                  float* __restrict__ Y) {
  const int lane = threadIdx.x & 31, lh = lane >> 4, ll = lane & 15;
  const int wid = blockIdx.x * 8 + (threadIdx.x >> 5);
  const int nColT = TOTn / 16;
  const int colBase = (wid % nColT) * 16;
  const int oBase = (wid / nColT) * 16;
  v8f acc = wmma_tile(W, X, oBase, colBase, CIpad, ll, lh);
  for (int v = 0; v < 8; ++v) {
    int ch = oBase + v + 8 * lh;
    float y = fmaxf(fmaf(acc[v], scale[ch], shift[ch]), 0.0f);
    Y[(size_t)ch * TOTn + colBase + ll] = y;
  }
}

// Last layer: GEMM + normalize + ReLU fused with max-over-K.
// All 16 cols of a tile share one (b,s); in-wave max then one int atomicMax
// (valid since ReLU output >= 0; max is order-independent => deterministic).
__global__ __launch_bounds__(256) void gemm_maxout(const float* __restrict__ W,
                                                   const float* __restrict__ X,
                                                   int CIpad,
                                                   const float* __restrict__ scale,
                                                   const float* __restrict__ shift,
                                                   int* __restrict__ outFeat) {
  const int lane = threadIdx.x & 31, lh = lane >> 4, ll = lane & 15;
  const int wid = blockIdx.x * 8 + (threadIdx.x >> 5);
  const int nColT = TOTn / 16;
  const int colBase = (wid % nColT) * 16;
  const int oBase = (wid / nColT) * 16;
  v8f acc = wmma_tile(W, X, oBase, colBase, CIpad, ll, lh);
  int g = colBase >> 5;     // b*S + s (16-col tile sits inside one K-group)
  int b = g >> 10, s = g & 1023;
  for (int v = 0; v < 8; ++v) {
    int ch = oBase + v + 8 * lh;
    float y = fmaxf(fmaf(acc[v], scale[ch], shift[ch]), 0.0f);
    for (int off = 1; off < 16; off <<= 1) y = fmaxf(y, __shfl_xor(y, off, 32));
    if (ll == 0)
      atomicMax(outFeat + ((size_t)b * 128 + ch) * Sn + s, __float_as_int(y));
  }
}

// ---------------------------------------------------------------------------
extern "C" void kernel_launch(void* const* d_in, const int* in_sizes, int n_in,
                              void* d_out, int out_size, void* d_ws, size_t ws_size,
                              hipStream_t stream) {
  const float* xyz  = (const float*)d_in[0];
  const float* feat = (const float*)d_in[1];
  const float* W0   = (const float*)d_in[2];
  const float* g0   = (const float*)d_in[3];
  const float* be0  = (const float*)d_in[4];
  const float* W1   = (const float*)d_in[5];
  const float* g1   = (const float*)d_in[6];
  const float* be1  = (const float*)d_in[7];
  const float* W2   = (const float*)d_in[8];
  const float* g2   = (const float*)d_in[9];
  const float* be2  = (const float*)d_in[10];
  float* out = (float*)d_out;

  // workspace partition (all 256B aligned)
  char* ws = (char*)d_ws;
  size_t off = 0;
  auto alloc = [&](size_t bytes) -> char* {
    char* p = ws + off;
    off = (off + bytes + 255) & ~(size_t)255;
    return p;
  };
  int*   gidx  = (int*)  alloc((size_t)TOTn * 4);
  float* Wpad0 = (float*)alloc((size_t)64 * 68 * 4);
  float* scl   = (float*)alloc(128 * 4);
  float* shf   = (float*)alloc(128 * 4);
  float* partS = (float*)alloc((size_t)128 * (TOTn / 128) * 4);
  float* partQ = (float*)alloc((size_t)128 * (TOTn / 128) * 4);
  float* XA    = (float*)alloc((size_t)68 * TOTn * 4);   // X0 (later reused as X2)
  float* XB    = (float*)alloc((size_t)64 * TOTn * 4);   // X1
  (void)ws_size; (void)in_sizes; (void)n_in;

  // Zero output (max-out accumulates via atomicMax; also makes calls idempotent)
  (void)hipMemsetAsync(d_out, 0, (size_t)out_size * sizeof(float), stream);

  // Sampling + grouping
  fps_kernel<<<Bn, 1024, 0, stream>>>(xyz, out);
  ballquery_kernel<<<(Bn * Sn) / 256, 256, 0, stream>>>(xyz, out, gidx);
  build_x0<<<TOTn / 256, 256, 0, stream>>>(xyz, feat, out, gidx, XA);
  pad_w0<<<17, 256, 0, stream>>>(W0, Wpad0);

  const int nParts = TOTn / 128;     // 4096
  const int nColT  = TOTn / 16;      // 32768

  // Layer 0: 68 -> 64
  gemm_stats<<<nParts * (64 / 16) / 8, 256, 0, stream>>>(Wpad0, XA, 68, partS, partQ);
  reduce_stats<<<64, 256, 0, stream>>>(partS, partQ, g0, be0, scl, shf);
  gemm_apply<<<nColT * (64 / 16) / 8, 256, 0, stream>>>(Wpad0, XA, 68, scl, shf, XB);

  // Layer 1: 64 -> 64
  gemm_stats<<<nParts * (64 / 16) / 8, 256, 0, stream>>>(W1, XB, 64, partS, partQ);
  reduce_stats<<<64, 256, 0, stream>>>(partS, partQ, g1, be1, scl, shf);
  gemm_apply<<<nColT * (64 / 16) / 8, 256, 0, stream>>>(W1, XB, 64, scl, shf, XA);

  // Layer 2: 64 -> 128, fused max over K into d_out
  gemm_stats<<<nParts * (128 / 16) / 8, 256, 0, stream>>>(W2, XA, 64, partS, partQ);
  reduce_stats<<<128, 256, 0, stream>>>(partS, partQ, g2, be2, scl, shf);
  gemm_maxout<<<nColT * (128 / 16) / 8, 256, 0, stream>>>(
      W2, XA, 64, scl, shf, (int*)(out + NXYZ));
}